// SteinerTopo_25048249270842
// MI455X (gfx1250) — compile-verified
//
#include <hip/hip_runtime.h>

// True clang vector types (required for __builtin_nontemporal_* — HIP's
// int4/float4 are structs and would be rejected).
typedef int   v4i __attribute__((ext_vector_type(4)));
typedef float v4f __attribute__((ext_vector_type(4)));

// Dual gather: out_x[i] = pos_x[idx_x[i]], out_y[i] = pos_y[idx_y[i]].
// Streaming traffic (indices in, outputs out) is tagged non-temporal so it
// does not evict the 64MB `pos` working set from the 192MB L2; the random
// gather loads use default (RT) temporal policy and should hit L2.
__global__ __launch_bounds__(256) void SteinerTopo_gather_kernel(
    const float* __restrict__ pos_x,
    const float* __restrict__ pos_y,
    const int*   __restrict__ idx_x,
    const int*   __restrict__ idx_y,
    float*       __restrict__ out_x,
    float*       __restrict__ out_y,
    int n_vec4,
    int n_total)
{
    const v4i* __restrict__ ix4 = (const v4i*)idx_x;
    const v4i* __restrict__ iy4 = (const v4i*)idx_y;
    v4f*       __restrict__ ox4 = (v4f*)out_x;
    v4f*       __restrict__ oy4 = (v4f*)out_y;

    const int stride = (int)(gridDim.x * blockDim.x);
    const int tid    = (int)(blockIdx.x * blockDim.x + threadIdx.x);

    for (int i = tid; i < n_vec4; i += stride) {
        // gfx1250: global_prefetch_b8 the next iteration's index lines.
        const int inext = i + stride;
        if (inext < n_vec4) {
            __builtin_prefetch((const void*)&ix4[inext], 0, 1);
            __builtin_prefetch((const void*)&iy4[inext], 0, 1);
        }

        // Non-temporal B128 index loads (streaming, read-once).
        v4i xi = __builtin_nontemporal_load(&ix4[i]);
        v4i yi = __builtin_nontemporal_load(&iy4[i]);

        // 8 independent gathers -> deep MLP per wave; RT hint keeps pos in L2.
        v4f xv, yv;
        xv.x = pos_x[xi.x];
        xv.y = pos_x[xi.y];
        xv.z = pos_x[xi.z];
        xv.w = pos_x[xi.w];
        yv.x = pos_y[yi.x];
        yv.y = pos_y[yi.y];
        yv.z = pos_y[yi.z];
        yv.w = pos_y[yi.w];

        // Non-temporal B128 stores (streaming, write-once).
        __builtin_nontemporal_store(xv, &ox4[i]);
        __builtin_nontemporal_store(yv, &oy4[i]);
    }

    // Scalar tail for n_total % 4 (none for V = 16M, kept for safety).
    for (int i = n_vec4 * 4 + tid; i < n_total; i += stride) {
        float xs = pos_x[idx_x[i]];
        float ys = pos_y[idx_y[i]];
        __builtin_nontemporal_store(xs, &out_x[i]);
        __builtin_nontemporal_store(ys, &out_y[i]);
    }
}

extern "C" void kernel_launch(void* const* d_in, const int* in_sizes, int n_in,
                              void* d_out, int out_size, void* d_ws, size_t ws_size,
                              hipStream_t stream) {
    // Input order (reference setup_inputs):
    //   0: pos              float32[2*P]
    //   1: pin_relate_x     int32[V]
    //   2: pin_relate_y     int32[V]
    //   3: net_vertex_start int32[N+1]   (unused by the computation)
    //   4: num_vertices     int32[1]     (V, also available as in_sizes[1])
    const float* pos   = (const float*)d_in[0];
    const int*   idx_x = (const int*)d_in[1];
    const int*   idx_y = (const int*)d_in[2];

    const int P = in_sizes[0] / 2;
    const int V = in_sizes[1];

    const float* pos_x = pos;
    const float* pos_y = pos + P;

    // Output: [newx (V floats) | newy (V floats)] flat.
    float* out_x = (float*)d_out;
    float* out_y = out_x + V;

    const int n_vec4 = V / 4;

    // 2048 blocks * 256 threads = 524288 threads (8 wave32 per block);
    // ~7-8 grid-stride iterations over n_vec4 = 4M so the prefetch-ahead
    // covers most of the index stream.
    const int threads = 256;
    int blocks = 2048;
    const int max_blocks = (n_vec4 + threads - 1) / threads;
    if (blocks > max_blocks && max_blocks > 0) blocks = max_blocks;
    if (blocks < 1) blocks = 1;

    SteinerTopo_gather_kernel<<<blocks, threads, 0, stream>>>(
        pos_x, pos_y, idx_x, idx_y, out_x, out_y, n_vec4, V);
}